// CelltypeScaleLayer_29162827940274
// MI455X (gfx1250) — compile-verified
//
#include <hip/hip_runtime.h>
#include <stdint.h>

// Problem constants (from the reference).
#define NCT       50
#define M_PER_CT  2000
#define N_FEATURE 20000
#define N_CELL    512
#define R_TOTAL   (NCT * M_PER_CT)   // 100000 gathered rows

// Tiling: 64 rows x 64 cells per block, 256 threads (8 wave32).
#define TILE_R 64
#define TILE_C 64

typedef float f32x4 __attribute__((ext_vector_type(4)));

// out[(c0+c)*R_TOTAL + r] = x[idx[r]*N_CELL + c0+c] * weight[r / M_PER_CT]
//
// Load phase : TENSOR-free async DMA global->LDS (global_load_async_to_lds_b128,
//              ASYNCcnt), raw x rows, coalesced 256B per tile-row.
// Store phase: apply per-row weight, transpose out of LDS (XOR-swizzled,
//              bank-conflict-free), non-temporal b128 stores so the 205MB
//              output stream does not evict the L2-resident 41MB x tensor.
__global__ __launch_bounds__(256) void
CelltypeScaleLayer_gather_scale_T(const float* __restrict__ x,
                                  const float* __restrict__ weight,
                                  const int*  __restrict__ idx,
                                  float* __restrict__ out)
{
    __shared__ __align__(16) float tile[TILE_R * TILE_C]; // 16 KB, no pad (XOR swizzle)
    __shared__ float wrow[TILE_R];

    const int t  = (int)threadIdx.x;
    const int r0 = (int)blockIdx.x * TILE_R;   // gathered-row tile origin
    const int c0 = (int)blockIdx.y * TILE_C;   // cell-column tile origin

    // Stage per-row weights for this tile (tiles can straddle celltype bounds).
    if (t < TILE_R) {
        int r = r0 + t;
        wrow[t] = (r < R_TOTAL) ? weight[r / M_PER_CT] : 0.0f;
    }

    // Raw 32-bit LDS byte offset of the tile (flat LDS aperture: low 32 bits).
    const uint32_t lds_base = (uint32_t)(uintptr_t)(&tile[0]);

    // ---- Async load phase: 1024 b128 chunks, 4 per thread ----------------
    // chunk id q -> tile row rl = q>>4, 16B chunk ch = q&15.
    // LDS chunk position XOR-swizzled by (rl>>2)&15 so the transposed column
    // reads below are bank-conflict free (16B granularity keeps alignment).
#pragma unroll
    for (int i = 0; i < 4; ++i) {
        const int q  = i * 256 + t;
        const int rl = q >> 4;
        const int ch = q & 15;
        if (r0 + rl < R_TOTAL) {
            const int row = idx[r0 + rl];                 // gather index
            const uint32_t voff = (uint32_t)row * (uint32_t)(N_CELL * 4)
                                + (uint32_t)(c0 * 4)
                                + (uint32_t)(ch * 16);
            const int sch = ch ^ ((rl >> 2) & 15);        // swizzled chunk
            const uint32_t laddr = lds_base
                                 + (uint32_t)(rl * TILE_C * 4)
                                 + (uint32_t)(sch * 16);
            asm volatile("global_load_async_to_lds_b128 %0, %1, %2"
                         :: "v"(laddr), "v"(voff), "s"(x)
                         : "memory");
        }
    }
    // This wave's DMAs have landed in LDS...
    asm volatile("s_wait_asynccnt 0" ::: "memory");
    // ...and the barrier publishes every wave's tile slice.
    __syncthreads();

    // ---- Store phase: scale + transpose + NT b128 stores ------------------
    // Thread t: rb = t&15 covers rows 4rb..4rb+3, cbase = t>>4 walks columns.
    const int rb    = t & 15;
    const int cbase = t >> 4;
    const int rloc  = rb << 2;
    const int rglob = r0 + rloc;

    if (rglob + 3 < R_TOTAL) {
        const float w0 = wrow[rloc + 0];
        const float w1 = wrow[rloc + 1];
        const float w2 = wrow[rloc + 2];
        const float w3 = wrow[rloc + 3];
#pragma unroll
        for (int i = 0; i < 4; ++i) {
            const int c = cbase + i * 16;                       // 0..63
            // physical (swizzled) float column inside the tile row:
            // rows 4rb..4rb+3 share r>>2 == rb, so one swizzle for all 4.
            const int scol = ((((c >> 2) ^ rb) << 2) | (c & 3));
            f32x4 v;
            v.x = tile[(rloc + 0) * TILE_C + scol] * w0;
            v.y = tile[(rloc + 1) * TILE_C + scol] * w1;
            v.z = tile[(rloc + 2) * TILE_C + scol] * w2;
            v.w = tile[(rloc + 3) * TILE_C + scol] * w3;
            const size_t off = (size_t)(c0 + c) * (size_t)R_TOTAL
                             + (size_t)rglob;                   // 16B aligned
            __builtin_nontemporal_store(v, (f32x4*)(out + off));
        }
    }
}

extern "C" void kernel_launch(void* const* d_in, const int* in_sizes, int n_in,
                              void* d_out, int out_size, void* d_ws, size_t ws_size,
                              hipStream_t stream)
{
    (void)in_sizes; (void)n_in; (void)d_ws; (void)ws_size; (void)out_size;

    const float* x      = (const float*)d_in[0];  // [20000, 512] f32
    const float* weight = (const float*)d_in[1];  // [50] f32
    const int*   idx    = (const int*)d_in[2];    // [50, 2000] int (harness int contract)
    float*       out    = (float*)d_out;          // [512 * 100000] f32

    dim3 grid((R_TOTAL + TILE_R - 1) / TILE_R,    // 1563 row tiles
              N_CELL / TILE_C);                   // 8 column tiles
    CelltypeScaleLayer_gather_scale_T<<<grid, dim3(256), 0, stream>>>(x, weight, idx, out);
}